// Net_7919919694130
// MI455X (gfx1250) — compile-verified
//
#include <hip/hip_runtime.h>
#include <float.h>

#define NUM_SENSE 3
#define EMB_DIM   300
#define BATCH     4096
#define CTX       10
#define NEG       5
#define CH        75          // EMB_DIM / 4 : row = exactly 75 float4 chunks
#define TINY      1.17549435e-38f

typedef __attribute__((ext_vector_type(2))) float v2f;
typedef __attribute__((ext_vector_type(8))) float v8f;
typedef __attribute__((ext_vector_type(4))) int   v4i;

#if defined(__has_builtin)
#if __has_builtin(__builtin_amdgcn_global_load_async_to_lds_b128)
#define HAVE_ASYNC_LDS 1
#endif
#endif

typedef __attribute__((address_space(1))) v4i gas_v4i;   // global (AS1) 16B vector
typedef __attribute__((address_space(3))) v4i las_v4i;   // LDS (AS3) 16B vector

// 16B global -> LDS copy; async DMA path on gfx1250 when the builtin exists.
__device__ __forceinline__ void copy16_g2l(const float* __restrict__ g, float* l) {
#ifdef HAVE_ASYNC_LDS
    __builtin_amdgcn_global_load_async_to_lds_b128((gas_v4i*)g, (las_v4i*)l, 0, 0);
#else
    *(float4*)l = *(const float4*)g;
#endif
}

__device__ __forceinline__ void wait_async_done() {
#ifdef HAVE_ASYNC_LDS
#if __has_builtin(__builtin_amdgcn_s_wait_asynccnt)
    __builtin_amdgcn_s_wait_asynccnt(0);
#else
    asm volatile("s_wait_asynccnt 0x0" ::: "memory");
#endif
#endif
}

__device__ __forceinline__ float wave_reduce_sum(float v) {
    v += __shfl_xor(v, 16, 32);
    v += __shfl_xor(v, 8, 32);
    v += __shfl_xor(v, 4, 32);
    v += __shfl_xor(v, 2, 32);
    v += __shfl_xor(v, 1, 32);
    return v;
}

// One disambiguation step, lane-transposed:
//   phase A: lane r<30 owns row (c=r/3, s=r%3): dot = <disamb_row, ctx>  (ctx broadcast via LDS)
//   phase B: per-context softmax over 3 senses (through LDS, no shuffles)
//   phase C: lane-chunked weighted accumulate of sense rows (coalesced float4 loads)
__device__ __forceinline__ void disamb_step(float4 ctx[3],
                                            const int* idsL, float* ctxL,
                                            float* dotL, float* aL,
                                            const float* __restrict__ emb_weight,
                                            const float* __restrict__ disamb_weight,
                                            int lane) {
    // publish ctx vector to LDS (float4 chunks q = lane + 32j)
#pragma unroll
    for (int j = 0; j < 3; ++j) {
        int q = lane + 32 * j;
        if (q < CH) *(float4*)&ctxL[q * 4] = ctx[j];
    }
    __syncthreads();

    // phase A: per-lane row dot product (rows streamed as float4)
    const int r  = lane;
    const int rs = (r < CTX * NUM_SENSE) ? r : 0;
    const int c  = rs / NUM_SENSE;
    const int s  = rs - NUM_SENSE * c;
    const float4* __restrict__ drow =
        (const float4*)(disamb_weight + ((long)idsL[c] * NUM_SENSE + s) * EMB_DIM);
    float dot = 0.0f;
#pragma unroll 5
    for (int q = 0; q < CH; ++q) {
        float4 a  = drow[q];
        float4 cv = *(const float4*)&ctxL[q * 4];   // same address all lanes: LDS broadcast
        dot += a.x * cv.x + a.y * cv.y + a.z * cv.z + a.w * cv.w;
    }
    dotL[lane] = dot;
    __syncthreads();

    // phase B: softmax over the 3 senses of this lane's context word
    {
        int cg = NUM_SENSE * c;
        float d0 = dotL[cg], d1 = dotL[cg + 1], d2 = dotL[cg + 2];
        float mx = fmaxf(d0, fmaxf(d1, d2));
        float es = expf(d0 - mx) + expf(d1 - mx) + expf(d2 - mx);
        aL[lane] = expf(dot - mx) / es;
    }
    __syncthreads();

    // phase C: ctx' = (1/CTX) * sum_r a[r] * sense_row_r   (lane-chunked, coalesced)
    float4 acc[3];
#pragma unroll
    for (int j = 0; j < 3; ++j) { acc[j].x = acc[j].y = acc[j].z = acc[j].w = 0.0f; }
    for (int rr = 0; rr < CTX * NUM_SENSE; ++rr) {
        int cc = rr / NUM_SENSE, ss = rr - NUM_SENSE * cc;
        float ar = aL[rr];                           // LDS broadcast
        const float4* __restrict__ srow =
            (const float4*)(emb_weight + ((long)idsL[cc] * NUM_SENSE + ss) * EMB_DIM);
#pragma unroll
        for (int j = 0; j < 3; ++j) {
            int q = lane + 32 * j;
            if (q < CH) {
                float4 v = srow[q];
                acc[j].x += ar * v.x; acc[j].y += ar * v.y;
                acc[j].z += ar * v.z; acc[j].w += ar * v.w;
            }
        }
    }
#pragma unroll
    for (int j = 0; j < 3; ++j) {
        ctx[j].x = acc[j].x * (1.0f / CTX); ctx[j].y = acc[j].y * (1.0f / CTX);
        ctx[j].z = acc[j].z * (1.0f / CTX); ctx[j].w = acc[j].w * (1.0f / CTX);
    }
    __syncthreads();
}

__global__ __launch_bounds__(32) void mssg_loss_kernel(
        const int*   __restrict__ word_ids,
        const int*   __restrict__ context_ids,
        const int*   __restrict__ neg_ids,
        const float* __restrict__ emb_weight,
        const float* __restrict__ global_emb_weight,
        const float* __restrict__ disamb_weight,
        float*       __restrict__ ws_partials) {
    __shared__ __align__(16) float tgt[16 * EMB_DIM];   // 0..9 ctx rows, 10..14 neg rows, 15 zero
    __shared__ __align__(16) float senseL[NUM_SENSE * EMB_DIM];
    __shared__ __align__(16) float ctxL[EMB_DIM + 4];
    __shared__ float dotL[32];
    __shared__ float aL[32];
    __shared__ int   idsL[CTX];

    const int b = blockIdx.x;
    const int lane = threadIdx.x;
    const int wid = word_ids[b];

    if (lane < CTX) idsL[lane] = context_ids[b * CTX + lane];
    __syncthreads();

    // ---- stage rows into LDS with async global->LDS b128 DMA ----
    for (int i = lane; i < CTX * CH; i += 32) {
        int c = i / CH, q = i - c * CH;
        copy16_g2l(global_emb_weight + (long)idsL[c] * EMB_DIM + q * 4,
                   &tgt[c * EMB_DIM + q * 4]);
    }
    for (int i = lane; i < NEG * CH; i += 32) {
        int n = i / CH, q = i - n * CH;
        copy16_g2l(global_emb_weight + (long)neg_ids[b * NEG + n] * EMB_DIM + q * 4,
                   &tgt[(CTX + n) * EMB_DIM + q * 4]);
    }
    for (int i = lane; i < NUM_SENSE * CH; i += 32) {
        int s = i / CH, q = i - s * CH;
        copy16_g2l(emb_weight + ((long)wid * NUM_SENSE + s) * EMB_DIM + q * 4,
                   &senseL[s * EMB_DIM + q * 4]);
    }
    for (int d = lane; d < EMB_DIM; d += 32) tgt[15 * EMB_DIM + d] = 0.0f;
    wait_async_done();
    __syncthreads();

    // ---- ctx1 = mean over context rows (float4 chunks q = lane + 32j) ----
    float4 ctx[3];
#pragma unroll
    for (int j = 0; j < 3; ++j) {
        int q = lane + 32 * j;
        float4 s4; s4.x = s4.y = s4.z = s4.w = 0.0f;
        if (q < CH) {
#pragma unroll
            for (int c = 0; c < CTX; ++c) {
                float4 t = *(const float4*)&tgt[c * EMB_DIM + q * 4];
                s4.x += t.x; s4.y += t.y; s4.z += t.z; s4.w += t.w;
            }
        }
        ctx[j].x = s4.x * (1.0f / CTX); ctx[j].y = s4.y * (1.0f / CTX);
        ctx[j].z = s4.z * (1.0f / CTX); ctx[j].w = s4.w * (1.0f / CTX);
    }

    // ---- two disambiguation refinement steps (ctx1 -> ctx2 -> ctx3) ----
    disamb_step(ctx, idsL, ctxL, dotL, aL, emb_weight, disamb_weight, lane);
    disamb_step(ctx, idsL, ctxL, dotL, aL, emb_weight, disamb_weight, lane);

    // ---- alpha = softmax_s( dot(disamb_embs[wid,s], ctx3) ), lane-per-row ----
#pragma unroll
    for (int j = 0; j < 3; ++j) {
        int q = lane + 32 * j;
        if (q < CH) *(float4*)&ctxL[q * 4] = ctx[j];
    }
    __syncthreads();
    {
        int rs = (lane < NUM_SENSE) ? lane : 0;
        const float4* __restrict__ arow =
            (const float4*)(disamb_weight + ((long)wid * NUM_SENSE + rs) * EMB_DIM);
        float dot = 0.0f;
#pragma unroll 5
        for (int q = 0; q < CH; ++q) {
            float4 a  = arow[q];
            float4 cv = *(const float4*)&ctxL[q * 4];
            dot += a.x * cv.x + a.y * cv.y + a.z * cv.z + a.w * cv.w;
        }
        dotL[lane] = dot;
    }
    __syncthreads();
    float al[NUM_SENSE];
    {
        float d0 = dotL[0], d1 = dotL[1], d2 = dotL[2];
        float mx = fmaxf(d0, fmaxf(d1, d2));
        float e0 = expf(d0 - mx), e1 = expf(d1 - mx), e2 = expf(d2 - mx);
        float inv = 1.0f / (e0 + e1 + e2);
        al[0] = e0 * inv; al[1] = e1 * inv; al[2] = e2 * inv;
    }

    // ---- scores z[s][t] = sense[3x300] x targets[300x16] via V_WMMA_F32_16X16X4_F32 ----
    // A (16x4 MxK): lane&15 = M, half-wave selects K pair; B (4x16 KxN): lane&15 = N.
    const int mrow  = lane & 15;
    const int khalf = (lane >> 4) * 2;
    const int msafe = (mrow < NUM_SENSE) ? mrow : 0;
    v8f acc = {};
    for (int k0 = 0; k0 < EMB_DIM; k0 += 4) {
        const int kb = k0 + khalf;
        v2f A, B;
        float ax = senseL[msafe * EMB_DIM + kb];
        float ay = senseL[msafe * EMB_DIM + kb + 1];
        A.x = (mrow < NUM_SENSE) ? ax : 0.0f;
        A.y = (mrow < NUM_SENSE) ? ay : 0.0f;
        B.x = tgt[mrow * EMB_DIM + kb];
        B.y = tgt[mrow * EMB_DIM + kb + 1];
        acc = __builtin_amdgcn_wmma_f32_16x16x4_f32(
                  false, A, false, B, (short)0, acc, false, false);
    }

    // ---- D layout: VGPR v holds M = v + 8*(lane>>4), N = lane&15; lanes 0..14 used ----
    float contrib = 0.0f;
    if (lane < CTX) {
        float p = al[0] / (1.0f + expf(-acc[0]))
                + al[1] / (1.0f + expf(-acc[1]))
                + al[2] / (1.0f + expf(-acc[2]));
        contrib = -logf(fmaxf(p, TINY));
    } else if (lane < CTX + NEG) {
        float q = al[0] / (1.0f + expf(-acc[0]))
                + al[1] / (1.0f + expf(-acc[1]))
                + al[2] / (1.0f + expf(-acc[2]));
        contrib = -logf(fmaxf(1.0f - q, TINY));
    }
    contrib = wave_reduce_sum(contrib);
    if (lane == 0) ws_partials[b] = contrib;
}

// Deterministic fixed-order final reduction: loss = sum(partials) / (BATCH*CTX)
__global__ __launch_bounds__(256) void mssg_reduce_kernel(
        const float* __restrict__ ws_partials, float* __restrict__ out) {
    __shared__ float sm[256];
    int t = threadIdx.x;
    float a = 0.0f;
    for (int i = t; i < BATCH; i += 256) a += ws_partials[i];
    sm[t] = a;
    __syncthreads();
    for (int w = 128; w > 0; w >>= 1) {
        if (t < w) sm[t] += sm[t + w];
        __syncthreads();
    }
    if (t == 0) out[0] = sm[0] * (1.0f / ((float)BATCH * (float)CTX));
}

extern "C" void kernel_launch(void* const* d_in, const int* in_sizes, int n_in,
                              void* d_out, int out_size, void* d_ws, size_t ws_size,
                              hipStream_t stream) {
    const int*   word_ids    = (const int*)d_in[0];
    const int*   context_ids = (const int*)d_in[1];
    // d_in[2] = context_masks (all ones, unused by the reference math)
    const int*   neg_ids     = (const int*)d_in[3];
    const float* emb_weight  = (const float*)d_in[4];
    const float* global_emb  = (const float*)d_in[5];
    const float* disamb      = (const float*)d_in[6];
    float*       out         = (float*)d_out;
    float*       partials    = (float*)d_ws;   // BATCH floats = 16 KB

    mssg_loss_kernel<<<BATCH, 32, 0, stream>>>(
        word_ids, context_ids, neg_ids, emb_weight, global_emb, disamb, partials);
    mssg_reduce_kernel<<<1, 256, 0, stream>>>(partials, out);
}